// BriLLMNodeBias_49435073577714
// MI455X (gfx1250) — compile-verified
//
#include <hip/hip_runtime.h>

#define B_ 64
#define T_ 512
#define D_ 32
#define V_ 16000
#define NSTEPS (T_ - 1)   // 511 scan steps
#define STAGES 8          // gather pipeline depth (LDS: 8*(4KB + 128B) ~ 33KB)

typedef float v2f __attribute__((ext_vector_type(2)));
typedef float v8f __attribute__((ext_vector_type(8)));

// ---------------------------------------------------------------------------
// Kernel A: per-batch sequential scan with an 8-stage pipelined random gather
// of W_pool[eid] (4KB/step) into LDS. One block per batch element; wave 0 runs
// the dependence chain (lane i owns e[i]); all 8 waves stream the gather.
// ---------------------------------------------------------------------------
__global__ __launch_bounds__(256) void brillm_scan_kernel(
    const int*   __restrict__ ids,
    const int*   __restrict__ eids,
    const float* __restrict__ bias_table,
    const float* __restrict__ W_pool,
    const float* __restrict__ b_pool,
    const float* __restrict__ a,
    const float* __restrict__ gate_p,
    const float* __restrict__ pe_scale_p,
    const float* __restrict__ PE,
    float*       __restrict__ e_out)
{
  __shared__ float sW[STAGES][D_ * D_];  // column-major W tiles
  __shared__ float sb[STAGES][D_];

  const int b   = blockIdx.x;
  const int tid = threadIdx.x;
  const float gate = gate_p[0];
  const float pesc = pe_scale_p[0];

  // e0 = bias_table[ids[:,0]] + pe_scale * PE[0]   (wave 0, lane i = e[i])
  float e = 0.0f;
  if (tid < 32) {
    const int id0 = ids[b * T_];
    e = bias_table[(size_t)id0 * D_ + tid] + pesc * PE[tid];
  }

  const int ebase = b * (T_ - 1);

  // Prologue: issue gathers for stages 0..STAGES-1 into registers.
  float4 rW[STAGES];
  float  rb[STAGES];
#pragma unroll
  for (int s = 0; s < STAGES; ++s) {
    const int eid = eids[ebase + s];
    rW[s] = *(const float4*)(W_pool + (size_t)eid * (D_ * D_) + tid * 4);
    if (tid < 32) rb[s] = b_pool[(size_t)eid * D_ + tid];
  }

  for (int base = 0; base < NSTEPS; base += STAGES) {
#pragma unroll
    for (int s = 0; s < STAGES; ++s) {
      const int t = base + s;
      if (t < NSTEPS) {
        // Drain stage regs -> LDS, transposing to column-major so the matvec
        // reads sW[j*32 + lane] conflict-free.
        const float4 w4 = rW[s];
        const int m = tid * 4;
        sW[s][((m + 0) & 31) * D_ + ((m + 0) >> 5)] = w4.x;
        sW[s][((m + 1) & 31) * D_ + ((m + 1) >> 5)] = w4.y;
        sW[s][((m + 2) & 31) * D_ + ((m + 2) >> 5)] = w4.z;
        sW[s][((m + 3) & 31) * D_ + ((m + 3) >> 5)] = w4.w;
        if (tid < 32) sb[s][tid] = rb[s];

        // Issue the gather for stage t+STAGES (stays in flight ~8 steps).
        const int tn = t + STAGES;
        if (tn < NSTEPS) {
          const int eidn = eids[ebase + tn];
          rW[s] = *(const float4*)(W_pool + (size_t)eidn * (D_ * D_) + tid * 4);
          if (tid < 32) rb[s] = b_pool[(size_t)eidn * D_ + tid];
        }
        // Warm L2 two pipeline depths ahead (global_prefetch_b8).
        const int tp = t + 2 * STAGES;
        if (tp < NSTEPS) {
          const int eidp = eids[ebase + tp];
          __builtin_prefetch(W_pool + (size_t)eidp * (D_ * D_) + tid * 4, 0, 1);
        }

        __syncthreads();  // stage s visible in LDS; loads keep flying

        if (tid < 32) {
          // h_i = tanh( sum_j W[i][j] * e_j + b_i )
          float acc = sb[s][tid];
#pragma unroll
          for (int j = 0; j < D_; ++j)
            acc = fmaf(sW[s][j * D_ + tid], __shfl(e, j), acc);
          const float ex = __expf(2.0f * acc);
          const float h  = 1.0f - 2.0f / (ex + 1.0f);   // tanh, saturates ±1
          const float at = a[t + 1];
          e = at * (gate * h + (1.0f - gate) * e) + pesc * PE[(t + 1) * D_ + tid];
        }
      }
    }
  }

  if (tid < 32) e_out[b * D_ + tid] = e;
}

// ---------------------------------------------------------------------------
// Kernel B: logits = e_final(64x32) @ bias_table^T(32x16000) via
// V_WMMA_F32_16X16X4_F32. One wave per 16-wide N strip; it computes all four
// 16-row M tiles, reusing the B fragment 4x per K step. 8 K-steps of 4.
// ---------------------------------------------------------------------------
__global__ __launch_bounds__(256) void brillm_logits_kernel(
    const float* __restrict__ ef,          // (64, 32) e_final
    const float* __restrict__ bias_table,  // (16000, 32)
    float*       __restrict__ out)         // (64, 16000)
{
  const int lane  = threadIdx.x & 31;
  const int wave  = threadIdx.x >> 5;
  const int ntile = blockIdx.x * 8 + wave;   // 0..999
  const int n0    = ntile * 16;
  const int l16   = lane & 15;
  const int half  = lane >> 4;               // lanes 16-31 hold K+2 / rows M+8

  v8f c0 = {}, c1 = {}, c2 = {}, c3 = {};
  const float* bn = bias_table + (size_t)(n0 + l16) * D_;  // row n of bias

#pragma unroll
  for (int ks = 0; ks < 8; ++ks) {
    const int kb = ks * 4 + half * 2;
    // B fragment: B[k][n] = bias_table[n][k]
    v2f bf;
    bf.x = bn[kb];
    bf.y = bn[kb + 1];
    // A fragments: four 16-row tiles of e_final, same K slice
    v2f a0, a1, a2, a3;
    a0.x = ef[(0  + l16) * D_ + kb]; a0.y = ef[(0  + l16) * D_ + kb + 1];
    a1.x = ef[(16 + l16) * D_ + kb]; a1.y = ef[(16 + l16) * D_ + kb + 1];
    a2.x = ef[(32 + l16) * D_ + kb]; a2.y = ef[(32 + l16) * D_ + kb + 1];
    a3.x = ef[(48 + l16) * D_ + kb]; a3.y = ef[(48 + l16) * D_ + kb + 1];

    c0 = __builtin_amdgcn_wmma_f32_16x16x4_f32(false, a0, false, bf, (short)0, c0, false, false);
    c1 = __builtin_amdgcn_wmma_f32_16x16x4_f32(false, a1, false, bf, (short)0, c1, false, false);
    c2 = __builtin_amdgcn_wmma_f32_16x16x4_f32(false, a2, false, bf, (short)0, c2, false, false);
    c3 = __builtin_amdgcn_wmma_f32_16x16x4_f32(false, a3, false, bf, (short)0, c3, false, false);
  }

  // C/D layout: VGPR r -> rows r (lanes 0-15) and r+8 (lanes 16-31), N = lane&15
  const int col = n0 + l16;
#pragma unroll
  for (int r = 0; r < 8; ++r) {
    const int row = r + half * 8;
    out[(size_t)(0  + row) * V_ + col] = c0[r];
    out[(size_t)(16 + row) * V_ + col] = c1[r];
    out[(size_t)(32 + row) * V_ + col] = c2[r];
    out[(size_t)(48 + row) * V_ + col] = c3[r];
  }
}

extern "C" void kernel_launch(void* const* d_in, const int* in_sizes, int n_in,
                              void* d_out, int out_size, void* d_ws, size_t ws_size,
                              hipStream_t stream) {
  const int*   ids  = (const int*)  d_in[0];
  const int*   eids = (const int*)  d_in[1];
  const float* bias = (const float*)d_in[2];
  const float* Wp   = (const float*)d_in[3];
  const float* bp   = (const float*)d_in[4];
  const float* a    = (const float*)d_in[5];
  const float* gate = (const float*)d_in[6];
  const float* pesc = (const float*)d_in[7];
  const float* PE   = (const float*)d_in[8];
  float* out = (float*)d_out;
  float* ef  = (float*)d_ws;  // 64*32 floats of scratch for e_final

  brillm_scan_kernel<<<B_, 256, 0, stream>>>(ids, eids, bias, Wp, bp, a,
                                             gate, pesc, PE, ef);
  brillm_logits_kernel<<<V_ / (16 * 8), 256, 0, stream>>>(ef, bias, out);
}